// CropSSIMLoss_57629871178353
// MI455X (gfx1250) — compile-verified
//
#include <hip/hip_runtime.h>

typedef __attribute__((ext_vector_type(2))) float v2f;
typedef __attribute__((ext_vector_type(8))) float v8f;

#define IMG_H 1080
#define IMG_W 1920
#define CROP  96
#define WIN   11
#define OUTD  86   // CROP - WIN + 1

__device__ __forceinline__ v8f wmma_f32(v2f a, v2f b, v8f c) {
  // D = A(16x4) * B(4x16) + C, fp32 matrix core op
  return __builtin_amdgcn_wmma_f32_16x16x4_f32(false, a, false, b, (short)0, c, false, false);
}

// One block = (crop n, channel ch, output row-block ty). 6 waves; wave w owns
// column tile x0 = 16*w for both the vertical (pass 1) and horizontal (pass 2)
// banded-matmul Gaussian filter passes.
__global__ __launch_bounds__(192) void ssim_tile_kernel(
    const float* __restrict__ img1, const float* __restrict__ img2,
    const int* __restrict__ target, float* __restrict__ partial) {
  __shared__ float Tq[5][16][100];   // stride 100 -> conflict-free strided reads
  __shared__ float gsh[WIN];
  __shared__ float wsum[6];

  const int tid  = threadIdx.x;
  const int lane = tid & 31;
  const int wid  = tid >> 5;      // 0..5
  const int l16  = lane & 15;
  const int half = lane >> 4;     // 0 or 1

  // Normalized Gaussian taps (sigma = 1.5): 2*sigma^2 = 4.5
  float gs = 0.f;
#pragma unroll
  for (int t = 0; t < WIN; ++t) { float cc = (float)t - 5.f; gs += expf(-cc * cc * (1.f / 4.5f)); }
  if (tid < WIN) { float cc = (float)tid - 5.f; gsh[tid] = expf(-cc * cc * (1.f / 4.5f)) / gs; }
  __syncthreads();

  const int b  = blockIdx.x;
  const int ty = b % 6;
  const int cn = b / 6;
  const int ch = cn % 3;
  const int n  = cn / 3;
  const int r0 = ty << 4;         // first output row of this block
  const int x0 = wid << 4;        // first column of this wave's tile

  const int xmin = target[n * 5 + 0];
  const int ymin = target[n * 5 + 1];
  const size_t plane = (size_t)ch * (size_t)(IMG_H * IMG_W);
  const float* p1 = img1 + plane;
  const float* p2 = img2 + plane;

  // Band weights per K-step. Identical formula serves as pass-1 A fragment
  // (Gv[r0+m, j], m = l16) and pass-2 B fragment (Gh[j, x0+i], i = l16):
  //   d = 4*s + 2*half + v - l16  (v = 0,1)
  float w0[7], w1[7];
#pragma unroll
  for (int s = 0; s < 7; ++s) {
    int d0 = 4 * s + 2 * half - l16;
    int d1 = d0 + 1;
    int c0 = min(max(d0, 0), WIN - 1);
    int c1 = min(max(d1, 0), WIN - 1);
    w0[s] = (d0 == c0) ? gsh[c0] : 0.f;
    w1[s] = (d1 == c1) ? gsh[c1] : 0.f;
  }

  // ---------------- Pass 1: vertical filter, T = Gv * Q ----------------
  v8f acc_a = {}, acc_b = {}, acc_aa = {}, acc_bb = {}, acc_ab = {};
  const int colg = xmin + x0 + l16;            // global image column (<= 1918)
#pragma unroll
  for (int s = 0; s < 7; ++s) {
    const int j0  = r0 + 4 * s + 2 * half;     // input row for fragment slot v=0
    const int jr0 = min(j0,     CROP - 1);     // clamp: weight is 0 beyond band
    const int jr1 = min(j0 + 1, CROP - 1);
    const float a0 = p1[(size_t)(ymin + jr0) * IMG_W + colg];
    const float a1 = p1[(size_t)(ymin + jr1) * IMG_W + colg];
    const float b0 = p2[(size_t)(ymin + jr0) * IMG_W + colg];
    const float b1 = p2[(size_t)(ymin + jr1) * IMG_W + colg];
    v2f af; af.x = w0[s]; af.y = w1[s];
    v2f q;
    q.x = a0;       q.y = a1;       acc_a  = wmma_f32(af, q, acc_a);
    q.x = b0;       q.y = b1;       acc_b  = wmma_f32(af, q, acc_b);
    q.x = a0 * a0;  q.y = a1 * a1;  acc_aa = wmma_f32(af, q, acc_aa);
    q.x = b0 * b0;  q.y = b1 * b1;  acc_bb = wmma_f32(af, q, acc_bb);
    q.x = a0 * b0;  q.y = a1 * b1;  acc_ab = wmma_f32(af, q, acc_ab);
  }

  // Spill T tiles to LDS (C/D layout: element (v + 8*half, l16))
#pragma unroll
  for (int v = 0; v < 8; ++v) {
    const int row = v + 8 * half;
    const int col = x0 + l16;
    Tq[0][row][col] = acc_a[v];
    Tq[1][row][col] = acc_b[v];
    Tq[2][row][col] = acc_aa[v];
    Tq[3][row][col] = acc_bb[v];
    Tq[4][row][col] = acc_ab[v];
  }
  __syncthreads();

  // ---------------- Pass 2: horizontal filter, S = T * Gh ----------------
  v8f mu1 = {}, mu2 = {}, fa2 = {}, fb2 = {}, fab = {};
  const int icol = x0 + l16;                          // output column of S
  const float bmask = (icol < OUTD) ? 1.f : 0.f;      // zero out cols >= 86
#pragma unroll
  for (int s = 0; s < 7; ++s) {
    const int j0  = x0 + 4 * s + 2 * half;
    const int jc0 = min(j0,     CROP - 1);            // weight 0 beyond band
    const int jc1 = min(j0 + 1, CROP - 1);
    v2f bf; bf.x = w0[s] * bmask; bf.y = w1[s] * bmask;
    v2f aq;
    aq.x = Tq[0][l16][jc0]; aq.y = Tq[0][l16][jc1];  mu1 = wmma_f32(aq, bf, mu1);
    aq.x = Tq[1][l16][jc0]; aq.y = Tq[1][l16][jc1];  mu2 = wmma_f32(aq, bf, mu2);
    aq.x = Tq[2][l16][jc0]; aq.y = Tq[2][l16][jc1];  fa2 = wmma_f32(aq, bf, fa2);
    aq.x = Tq[3][l16][jc0]; aq.y = Tq[3][l16][jc1];  fb2 = wmma_f32(aq, bf, fb2);
    aq.x = Tq[4][l16][jc0]; aq.y = Tq[4][l16][jc1];  fab = wmma_f32(aq, bf, fab);
  }

  // ---------------- SSIM map + masked reduction ----------------
  const float C1c = 6.5025f;    // (0.01*255)^2
  const float C2c = 58.5225f;   // (0.03*255)^2
  float lsum = 0.f;
#pragma unroll
  for (int v = 0; v < 8; ++v) {
    const int row = r0 + v + 8 * half;
    if (row < OUTD && icol < OUTD) {
      const float m1 = mu1[v], m2 = mu2[v];
      const float m1s = m1 * m1, m2s = m2 * m2, m12 = m1 * m2;
      const float s1  = fa2[v] - m1s;
      const float s2  = fb2[v] - m2s;
      const float s12 = fab[v] - m12;
      const float num = (2.f * m12 + C1c) * (2.f * s12 + C2c);
      const float den = (m1s + m2s + C1c) * (s1 + s2 + C2c);
      lsum += num / den;
    }
  }
#pragma unroll
  for (int off = 16; off > 0; off >>= 1)
    lsum += __shfl_xor(lsum, off, 32);
  if (lane == 0) wsum[wid] = lsum;
  __syncthreads();
  if (tid == 0) {
    float t = 0.f;
#pragma unroll
    for (int i = 0; i < 6; ++i) t += wsum[i];
    partial[b] = t;
  }
}

__global__ __launch_bounds__(256) void ssim_finalize_kernel(
    const float* __restrict__ partial, float* __restrict__ out,
    int nblocks, int ncrops) {
  __shared__ double sh[256];
  double s = 0.0;
  for (int i = threadIdx.x; i < nblocks; i += 256) s += (double)partial[i];
  sh[threadIdx.x] = s;
  __syncthreads();
  for (int st = 128; st > 0; st >>= 1) {
    if ((int)threadIdx.x < st) sh[threadIdx.x] += sh[threadIdx.x + st];
    __syncthreads();
  }
  if (threadIdx.x == 0) {
    const double denom = (double)ncrops * 3.0 * (double)OUTD * (double)OUTD;
    out[0] = (float)(5.0 - 5.0 * sh[0] / denom);
  }
}

extern "C" void kernel_launch(void* const* d_in, const int* in_sizes, int n_in,
                              void* d_out, int out_size, void* d_ws, size_t ws_size,
                              hipStream_t stream) {
  (void)n_in; (void)out_size; (void)ws_size;
  const float* image   = (const float*)d_in[0];
  const float* imageRW = (const float*)d_in[1];
  const int*   target  = (const int*)d_in[2];
  const int ncrops  = in_sizes[2] / 5;     // target is (N,5)
  const int nblocks = ncrops * 3 * 6;      // crop x channel x row-block
  float* partial = (float*)d_ws;

  ssim_tile_kernel<<<nblocks, 192, 0, stream>>>(image, imageRW, target, partial);
  ssim_finalize_kernel<<<1, 256, 0, stream>>>(partial, (float*)d_out, nblocks, ncrops);
}